// WarpingLayer_21715354649746
// MI455X (gfx1250) — compile-verified
//
#include <hip/hip_runtime.h>

// ---------------------------------------------------------------------------
// WarpingLayer (bilinear flow warp, zeros padding, align_corners=False mask)
// x:[B=8,C=96,H=128,W=256] f32, flow:[B,2,H,W] f32 -> out:[B,C,H,W] f32
// Bandwidth-bound gather: ~203 MB total traffic -> ~8.7us floor @ 23.3 TB/s.
// CDNA5-specific paths: global_load_async_to_lds_b32 (+ s_wait_asynccnt) for
// flow staging; non-temporal output stores keep x resident in the 192MB L2.
// Dims are compile-time constants -> pow2 shifts instead of sdiv sequences.
// ---------------------------------------------------------------------------

#define GAS __attribute__((address_space(1)))
#define LAS __attribute__((address_space(3)))

#if __has_builtin(__builtin_amdgcn_global_load_async_to_lds_b32)
#define HAVE_ASYNC_LDS 1
#else
#define HAVE_ASYNC_LDS 0
#endif

namespace {
constexpr int kB = 8;
constexpr int kC = 96;
constexpr int kH = 128;   // pow2: b/h split is shift/and
constexpr int kW = 256;   // == blockDim.x
constexpr size_t kHW = (size_t)kH * (size_t)kW;
}

__global__ __launch_bounds__(kW) void warp_bilinear_kernel(
    const float* __restrict__ x, const float* __restrict__ flow,
    const int* __restrict__ p_him, const int* __restrict__ p_wim,
    const int* __restrict__ p_div, float* __restrict__ out)
{
    __shared__ float lds_flow[2 * kW];  // [0..W-1]=flo_w row, [W..2W-1]=flo_h row

    const int w  = threadIdx.x;         // 0..kW-1
    const int bh = blockIdx.x;          // b*kH + h
    const int h  = bh & (kH - 1);
    const int b  = bh >> 7;             // kH == 128

    // ---- stage this row's flow through LDS via the gfx1250 async path ----
    const float* fwp = flow + ((size_t)b * 2) * kHW + (size_t)h * kW + w;
    const float* fhp = fwp + kHW;

    float fw, fh;
#if HAVE_ASYNC_LDS
    __builtin_amdgcn_global_load_async_to_lds_b32(
        (GAS int*)fwp, (LAS int*)&lds_flow[w], 0, 0);
    __builtin_amdgcn_global_load_async_to_lds_b32(
        (GAS int*)fhp, (LAS int*)&lds_flow[kW + w], 0, 0);
  #if __has_builtin(__builtin_amdgcn_s_wait_asynccnt)
    __builtin_amdgcn_s_wait_asynccnt(0);
  #else
    asm volatile("s_wait_asynccnt 0x0" ::: "memory");
  #endif
    asm volatile("" ::: "memory");      // order LDS reads after the wait
    fw = lds_flow[w];
    fh = lds_flow[kW + w];
#else
    fw = *fwp;
    fh = *fhp;
#endif

    // ---- per-pixel sample position (computed once, reused for 96 channels) --
    const int him = *p_him;
    const int wim = *p_wim;
    const float divf = (float)(*p_div);
    const int wden = (wim - 1) > 1 ? (wim - 1) : 1;
    const int hden = (him - 1) > 1 ? (him - 1) : 1;
    const float sW = (float)(2.0 / (double)wden / (double)divf);
    const float sH = (float)(2.0 / (double)hden / (double)divf);

    const float gx = -1.0f + (2.0f / (float)(kW - 1)) * (float)w + fw * sW;
    const float gy = -1.0f + (2.0f / (float)(kH - 1)) * (float)h + fh * sH;

    const float ix = ((gx + 1.0f) * (float)kW - 1.0f) * 0.5f;
    const float iy = ((gy + 1.0f) * (float)kH - 1.0f) * 0.5f;

    const float x0f = floorf(ix);
    const float y0f = floorf(iy);
    const int x0 = (int)x0f;
    const int y0 = (int)y0f;
    const float ax = ix - x0f;          // in [0,1)
    const float ay = iy - y0f;

    // per-corner validity (zeros padding)
    const bool vx0 = (x0 >= 0) && (x0 < kW);
    const bool vx1 = (x0 + 1 >= 0) && (x0 + 1 < kW);
    const bool vy0 = (y0 >= 0) && (y0 < kH);
    const bool vy1 = (y0 + 1 >= 0) && (y0 + 1 < kH);

    float wv00 = (1.0f - ax) * (1.0f - ay) * ((vx0 && vy0) ? 1.0f : 0.0f);
    float wv10 = ax * (1.0f - ay)          * ((vx1 && vy0) ? 1.0f : 0.0f);
    float wv01 = (1.0f - ax) * ay          * ((vx0 && vy1) ? 1.0f : 0.0f);
    float wv11 = ax * ay                   * ((vx1 && vy1) ? 1.0f : 0.0f);

    // mask = (grid_sample(ones) >= 1), accumulated in reference corner order
    const float mval = ((wv00 + wv10) + wv01) + wv11;
    const float mask = (mval >= 1.0f) ? 1.0f : 0.0f;
    wv00 *= mask; wv10 *= mask; wv01 *= mask; wv11 *= mask;

    // clamped flat offsets within one [H,W] channel image
    const int xi0 = x0     < 0 ? 0 : (x0     > kW - 1 ? kW - 1 : x0);
    const int xi1 = x0 + 1 < 0 ? 0 : (x0 + 1 > kW - 1 ? kW - 1 : x0 + 1);
    const int yi0 = y0     < 0 ? 0 : (y0     > kH - 1 ? kH - 1 : y0);
    const int yi1 = y0 + 1 < 0 ? 0 : (y0 + 1 > kH - 1 ? kH - 1 : y0 + 1);
    const int off00 = (yi0 << 8) + xi0;   // kW == 256
    const int off10 = (yi0 << 8) + xi1;
    const int off01 = (yi1 << 8) + xi0;
    const int off11 = (yi1 << 8) + xi1;

    // ---- channel loop: 4 gathers (L2-resident x) + NT store per channel ----
    const float* xb = x + (size_t)b * (size_t)kC * kHW;
    float* ob = out + (size_t)b * (size_t)kC * kHW + (size_t)h * kW + w;

#pragma unroll 8
    for (int c = 0; c < kC; ++c) {
        const float* pc = xb + (size_t)c * kHW;
        const float g00 = pc[off00];
        const float g10 = pc[off10];
        const float g01 = pc[off01];
        const float g11 = pc[off11];
        const float r = ((g00 * wv00 + g10 * wv10) + g01 * wv01) + g11 * wv11;
        // non-temporal: output is write-once; keep x resident in the 192MB L2
        __builtin_nontemporal_store(r, ob + (size_t)c * kHW);
    }
}

extern "C" void kernel_launch(void* const* d_in, const int* in_sizes, int n_in,
                              void* d_out, int out_size, void* d_ws, size_t ws_size,
                              hipStream_t stream) {
    const float* x    = (const float*)d_in[0];
    const float* flow = (const float*)d_in[1];
    const int* him    = (const int*)d_in[2];
    const int* wim    = (const int*)d_in[3];
    const int* divf   = (const int*)d_in[4];
    float* out        = (float*)d_out;

    // one block per (b,h) row; 256 threads = 8 wave32 per block
    dim3 grid(kB * kH), block(kW);
    hipLaunchKernelGGL(warp_bilinear_kernel, grid, block, 0, stream,
                       x, flow, him, wim, divf, out);
}